// TCPLoss_49658411877125
// MI455X (gfx1250) — compile-verified
//
#include <hip/hip_runtime.h>
#include <stdint.h>

#define B_DIM 4096
#define C_DIM 50257
#define TPB 256
#define TILE_FLOATS 1024            // TPB * 4 floats per tile
#define TILE_BYTES  4096
#define NTILES 49                   // 49 * 1024 = 50176 bulk floats
#define BULK (NTILES * TILE_FLOATS)
#define LOG2E  1.4426950408889634f
#define LN2    0.6931471805599453f
#define NEG_INF (-3.402823466e38f)

__global__ __launch_bounds__(TPB) void tcp_row_loss_kernel(
    const float* __restrict__ pred,
    const long long* __restrict__ targets,
    float* __restrict__ row_terms)
{
    __shared__ __align__(16) float buf[2][TILE_FLOATS];   // async double buffer (8 KB)
    __shared__ float red_m[TPB / 32];
    __shared__ float red_s[TPB / 32];

    const int row  = blockIdx.x;
    const int tid  = threadIdx.x;
    const int lane = tid & 31;
    const int wid  = tid >> 5;

    const size_t base_elt = (size_t)row * (size_t)C_DIM;
    const float* rp  = pred + base_elt;                 // row start (4B aligned only)
    const int h      = (int)((4u - ((uint32_t)base_elt & 3u)) & 3u); // peel to 16B align
    const float* rpa = rp + h;                          // 16B-aligned bulk base
    const int tailn  = C_DIM - h - BULK;                // 78..81 tail floats

    // ---- online softmax state (base-2): s = sum exp2((x - m) * log2e) ----
    float m = NEG_INF;
    float s = 0.0f;

    auto push1 = [&](float x) {
        const float nm = fmaxf(m, x);
        s = s * __builtin_amdgcn_exp2f((m - nm) * LOG2E)
          + __builtin_amdgcn_exp2f((x - nm) * LOG2E);
        m = nm;
    };

    // head (unaligned prefix) + tail handled with plain scalar loads
    if (tid < h)     push1(rp[tid]);
    if (tid < tailn) push1(rpa[BULK + tid]);

    // per-lane LDS byte addresses for the two buffers (generic ptr low 32 bits == LDS offset)
    const uint32_t lds0 = (uint32_t)(uintptr_t)(&buf[0][tid << 2]);
    const uint32_t lds1 = (uint32_t)(uintptr_t)(&buf[1][tid << 2]);
    const uint32_t goff0 = (uint32_t)(tid << 4);        // byte offset of this lane in tile 0

    // prologue: kick off tile 0 DMA into buf[0]
    asm volatile("global_load_async_to_lds_b128 %0, %1, %2"
                 :: "v"(lds0), "v"(goff0), "s"(rpa) : "memory");

    for (int t = 0; t < NTILES; ++t) {
        if (t + 1 < NTILES) {
            const uint32_t nxt_lds = ((t + 1) & 1) ? lds1 : lds0;
            const uint32_t nxt_off = goff0 + (uint32_t)(t + 1) * TILE_BYTES;
            // buffer about to be overwritten was read in the previous iter; make sure those
            // ds_loads have fully retired before the async engine writes it again
            asm volatile("s_wait_dscnt 0x0" ::: "memory");
            asm volatile("global_load_async_to_lds_b128 %0, %1, %2"
                         :: "v"(nxt_lds), "v"(nxt_off), "s"(rpa) : "memory");
            asm volatile("s_wait_asynccnt 0x1" ::: "memory"); // tile t landed; t+1 in flight
        } else {
            asm volatile("s_wait_asynccnt 0x0" ::: "memory"); // last tile landed
        }

        const float4 v = *(const float4*)((t & 1) ? &buf[1][tid << 2] : &buf[0][tid << 2]);

        const float a = fmaxf(fmaxf(v.x, v.y), fmaxf(v.z, v.w));
        if (a > m) {                      // rare after warm-up -> uniform skip
            s *= __builtin_amdgcn_exp2f((m - a) * LOG2E);
            m = a;
        }
        const float c = m * LOG2E;
        s += __builtin_amdgcn_exp2f(__builtin_fmaf(v.x, LOG2E, -c))
           + __builtin_amdgcn_exp2f(__builtin_fmaf(v.y, LOG2E, -c))
           + __builtin_amdgcn_exp2f(__builtin_fmaf(v.z, LOG2E, -c))
           + __builtin_amdgcn_exp2f(__builtin_fmaf(v.w, LOG2E, -c));
    }

    // ---- intra-wave merge (wave32: 5 xor levels) ----
    #pragma unroll
    for (int off = 16; off > 0; off >>= 1) {
        const float om = __shfl_xor(m, off, 32);
        const float os = __shfl_xor(s, off, 32);
        const float nm = fmaxf(m, om);
        s = s  * __builtin_amdgcn_exp2f((m  - nm) * LOG2E)
          + os * __builtin_amdgcn_exp2f((om - nm) * LOG2E);
        m = nm;
    }

    if (lane == 0) { red_m[wid] = m; red_s[wid] = s; }
    __syncthreads();

    // ---- cross-wave merge on wave 0 (8 partials) ----
    if (wid == 0) {
        float mm = (lane < (TPB / 32)) ? red_m[lane] : NEG_INF;
        float ss = (lane < (TPB / 32)) ? red_s[lane] : 0.0f;
        #pragma unroll
        for (int off = 4; off > 0; off >>= 1) {
            const float om = __shfl_xor(mm, off, 32);
            const float os = __shfl_xor(ss, off, 32);
            const float nm = fmaxf(mm, om);
            ss = ss * __builtin_amdgcn_exp2f((mm - nm) * LOG2E)
               + os * __builtin_amdgcn_exp2f((om - nm) * LOG2E);
            mm = nm;
        }
        if (lane == 0) {
            const long long tg = targets[row];
            const float xt = rp[tg];                               // L2 hit: row just streamed
            const float logpt = (xt - mm) - __builtin_amdgcn_logf(ss) * LN2; // ln(sum)
            const float p  = __builtin_amdgcn_exp2f(logpt * LOG2E);
            const float ce = -logpt;
            row_terms[row] = (2.0f - 2.0f * p) * ce;
        }
    }
}

__global__ __launch_bounds__(TPB) void tcp_mean_kernel(
    const float* __restrict__ row_terms, float* __restrict__ out)
{
    __shared__ float red[TPB / 32];
    const int tid  = threadIdx.x;
    const int lane = tid & 31;
    const int wid  = tid >> 5;

    float s = 0.0f;
    #pragma unroll
    for (int i = 0; i < B_DIM / TPB; ++i)
        s += row_terms[tid + i * TPB];

    #pragma unroll
    for (int off = 16; off > 0; off >>= 1)
        s += __shfl_xor(s, off, 32);

    if (lane == 0) red[wid] = s;
    __syncthreads();

    if (tid == 0) {
        float t = 0.0f;
        #pragma unroll
        for (int i = 0; i < TPB / 32; ++i) t += red[i];
        out[0] = t * (1.0f / (float)B_DIM);
    }
}

extern "C" void kernel_launch(void* const* d_in, const int* in_sizes, int n_in,
                              void* d_out, int out_size, void* d_ws, size_t ws_size,
                              hipStream_t stream) {
    const float*     pred    = (const float*)d_in[0];
    const long long* targets = (const long long*)d_in[1];
    float* row_terms = (float*)d_ws;        // 4096 floats = 16 KB scratch
    float* out       = (float*)d_out;

    tcp_row_loss_kernel<<<B_DIM, TPB, 0, stream>>>(pred, targets, row_terms);
    tcp_mean_kernel<<<1, TPB, 0, stream>>>(row_terms, out);
}